// CayleyCirculantSSMLayer_62302795596462
// MI455X (gfx1250) — compile-verified
//
#include <hip/hip_runtime.h>
#include <hip/hip_bf16.h>

// Cayley-circulant SSM on MI455X (gfx1250, wave32).
//   X = u @ (F B_w)^T                    (WMMA f32 GEMM, M=16384 K=1024 N=1152)
//   hhat_t = lambda * hhat_{t-1} + X_t   (independent per (batch, mode) scan)
//   y = [Re|Im hhat] @ (C_w F^-1)^T + D*u (WMMA f32 GEMM, K=1152 N=1024)
// GEMMs use double-buffered GLOBAL_LOAD_ASYNC_TO_LDS staging (ASYNCcnt) so the
// tile-(k+1) DMA overlaps tile-k v_wmma issue.

#define BATCH 8
#define SEQ 2048
#define DM 1024
#define NST 1024
#define MTOT (BATCH * SEQ)       // 16384
#define NKP 576                  // padded mode count (modes 513..575 are zero)
#define KX (2 * NKP)             // 1152: [re modes | im modes]

typedef float v2f __attribute__((ext_vector_type(2)));
typedef float v8f __attribute__((ext_vector_type(8)));
typedef int v2i_vs __attribute__((vector_size(8)));   // matches builtin param pointee

// ---------------- async global->LDS copy (CDNA5) ----------------
__device__ __forceinline__ void async_copy_b64(const float* __restrict__ g, float* lds) {
#if __has_builtin(__builtin_amdgcn_global_load_async_to_lds_b64)
    __builtin_amdgcn_global_load_async_to_lds_b64(
        (__attribute__((address_space(1))) v2i_vs*)g,
        (__attribute__((address_space(3))) v2i_vs*)lds, 0, 0);
#else
    unsigned off = (unsigned)(unsigned long long)(__attribute__((address_space(3))) float*)lds;
    asm volatile("global_load_async_to_lds_b64 %0, %1, off" ::"v"(off), "v"(g) : "memory");
#endif
}

__device__ __forceinline__ void wait_async_le16() {
#if __has_builtin(__builtin_amdgcn_s_wait_asynccnt)
    __builtin_amdgcn_s_wait_asynccnt(16);
#else
    asm volatile("s_wait_asynccnt 0x10" ::: "memory");
#endif
}

__device__ __forceinline__ void wait_async_le0() {
#if __has_builtin(__builtin_amdgcn_s_wait_asynccnt)
    __builtin_amdgcn_s_wait_asynccnt(0);
#else
    asm volatile("s_wait_asynccnt 0x0" ::: "memory");
#endif
}

// ---------------- trig table: cos/sin(2*pi*j/1024), j=0..1023 ----------------
__global__ void trig_kernel(float* __restrict__ trig) {
    int j = blockIdx.x * 256 + threadIdx.x;
    if (j < 1024) {
        float s, c;
        sincosf(6.28318530717958647692f * (float)j / 1024.0f, &s, &c);
        trig[j]        = c;
        trig[1024 + j] = s;
    }
}

// ---------------- lambda_k = Cayley(i*omega_k) for k=0..512; 0 for pad ----------------
__global__ void lambda_kernel(const float* __restrict__ a, const float* __restrict__ trig,
                              float* __restrict__ lre, float* __restrict__ lim) {
    int k = blockIdx.x * 64 + threadIdx.x;
    if (k >= NKP) return;
    if (k > 512) { lre[k] = 0.0f; lim[k] = 0.0f; return; }
    const float* sinT = trig + 1024;
    float w = 0.0f;
    int idx = 0;
    for (int j = 1; j <= 511; ++j) {
        idx = (idx + k) & 1023;                 // j*k mod 1024
        w = fmaf(a[j - 1], sinT[idx], w);
    }
    w = -2.0f * w;
    float w2 = w * w, den = 1.0f + w2;
    lre[k] = (1.0f - w2) / den;
    lim[k] = -2.0f * w / den;
}

// ---------------- FB[k'][d]: rows 0..575 = sum_j cos(2pi k j/n) B[j][d]; 576.. = -sin ----------------
__global__ __launch_bounds__(256) void fb_kernel(const float* __restrict__ Bw,
                                                 const float* __restrict__ trig,
                                                 float* __restrict__ FB) {
    __shared__ float tab[1024];
    const int kp0 = blockIdx.x * 8;             // 144 blocks x 8 rows
    const bool isIm = kp0 >= NKP;
    const int kbase = isIm ? kp0 - NKP : kp0;
    const float* g = trig + (isIm ? 1024 : 0);
    for (int i = threadIdx.x; i < 1024; i += 256) tab[i] = g[i];
    __syncthreads();

    float acc[8][4] = {};
    const int d0 = threadIdx.x;
    for (int j = 0; j < 1024; ++j) {
        float bw[4];
#pragma unroll
        for (int i = 0; i < 4; ++i) bw[i] = Bw[j * DM + d0 + i * 256];
#pragma unroll
        for (int r = 0; r < 8; ++r) {
            float t = tab[(j * (kbase + r)) & 1023];   // uniform LDS broadcast
#pragma unroll
            for (int i = 0; i < 4; ++i) acc[r][i] = fmaf(t, bw[i], acc[r][i]);
        }
    }
#pragma unroll
    for (int r = 0; r < 8; ++r) {
        int k = kbase + r;
        float scl = (k <= 512) ? (isIm ? -1.0f : 1.0f) : 0.0f;
#pragma unroll
        for (int i = 0; i < 4; ++i)
            FB[(size_t)(kp0 + r) * DM + d0 + i * 256] = acc[r][i] * scl;
    }
}

// ---------------- Wc[d][k']: (c_k/n)*sum_j C[d][j]cos(..) ; -(c_k/n)*sum_j C[d][j]sin(..) ----------------
__global__ __launch_bounds__(288) void cf_kernel(const float* __restrict__ Cw,
                                                 const float* __restrict__ trig,
                                                 float* __restrict__ Wc) {
    __shared__ float cosT[1024];
    __shared__ float sinT[1024];
    __shared__ float crow[1024];
    const int d = blockIdx.x;
    for (int i = threadIdx.x; i < 1024; i += 288) {
        cosT[i] = trig[i];
        sinT[i] = trig[1024 + i];
        crow[i] = Cw[(size_t)d * NST + i];
    }
    __syncthreads();

    int kp[4], km[4];
    const float* tp[4];
#pragma unroll
    for (int i = 0; i < 4; ++i) {
        kp[i] = threadIdx.x + i * 288;
        bool isIm = kp[i] >= NKP;
        km[i] = isIm ? kp[i] - NKP : kp[i];
        tp[i] = isIm ? sinT : cosT;
    }
    float acc[4] = {};
    for (int j = 0; j < 1024; ++j) {
        float c = crow[j];
#pragma unroll
        for (int i = 0; i < 4; ++i)
            acc[i] = fmaf(c, tp[i][(j * km[i]) & 1023], acc[i]);
    }
#pragma unroll
    for (int i = 0; i < 4; ++i) {
        int k = km[i];
        float ck = (k > 512) ? 0.0f : ((k == 0 || k == 512) ? 1.0f : 2.0f);
        float scl = ck * (1.0f / 1024.0f) * ((kp[i] >= NKP) ? -1.0f : 1.0f);
        Wc[(size_t)d * KX + kp[i]] = acc[i] * scl;
    }
}

// ---------------- per-mode complex recurrence (in place over X) ----------------
__global__ void scan_kernel(float* __restrict__ X, const float* __restrict__ lre,
                            const float* __restrict__ lim) {
    int t = blockIdx.x * 256 + threadIdx.x;
    if (t >= BATCH * NKP) return;
    const int k = t % NKP, b = t / NKP;
    const float lr = lre[k], li = lim[k];
    float hr = 0.0f, hi = 0.0f;
    float* p = X + (size_t)b * SEQ * KX + k;
    for (int s = 0; s < SEQ; ++s) {
        float xr = p[0], xi = p[NKP];
        float nr = fmaf(lr, hr, fmaf(-li, hi, xr));
        float ni = fmaf(lr, hi, fmaf(li, hr, xi));
        p[0] = nr; p[NKP] = ni;
        hr = nr; hi = ni;
        p += KX;
    }
}

// ---------------- f32 WMMA GEMM:  C[m][n] = sum_k A[m][k] * W[n][k]  (+ D[n]*U[m][n]) ----------------
// Block tile 128(M) x 128(N), BK=32. 8 waves: 4(M) x 2(N) grid, wave tile 32x64 = 2x4 accums.
// Double-buffered async global->LDS staging: 16 b64 transfers per wave per tile.
template <bool EPI>
__global__ __launch_bounds__(256) void wmma_gemm(const float* __restrict__ A,
                                                 const float* __restrict__ W,
                                                 float* __restrict__ C, int K, int lda,
                                                 int ldw, int ldc,
                                                 const float* __restrict__ U,
                                                 const float* __restrict__ Dv) {
    constexpr int LS = 34;                      // conflict-free LDS row stride (floats)
    __shared__ __align__(16) float As[2][128 * LS];
    __shared__ __align__(16) float Bs[2][128 * LS];

    const int tid = threadIdx.x;
    const int wid = tid >> 5, lane = tid & 31;
    const int lm = lane & 15, lh = lane >> 4;
    const int waveM = wid & 3, waveN = wid >> 2;
    const int m0 = blockIdx.y * 128;
    const int n0 = blockIdx.x * 128;

    // per-thread staging coordinates (8 float2 each for A and W per tile)
    const int srow = tid >> 4;                  // +16 per i -> rows 0..127
    const int scp = (tid & 15) << 1;

    v8f acc[2][4] = {};

    const int ntiles = K >> 5;

    // prologue: stage tile 0 into buffer 0
#pragma unroll
    for (int i = 0; i < 8; ++i) {
        int row = srow + i * 16;
        async_copy_b64(A + (size_t)(m0 + row) * lda + scp, &As[0][row * LS + scp]);
        async_copy_b64(W + (size_t)(n0 + row) * ldw + scp, &Bs[0][row * LS + scp]);
    }

    for (int kt = 0; kt < ntiles; ++kt) {
        const int cur = kt & 1;
        if (kt + 1 < ntiles) {
            const int kof = (kt + 1) << 5;
#pragma unroll
            for (int i = 0; i < 8; ++i) {
                int row = srow + i * 16;
                async_copy_b64(A + (size_t)(m0 + row) * lda + kof + scp,
                               &As[cur ^ 1][row * LS + scp]);
                async_copy_b64(W + (size_t)(n0 + row) * ldw + kof + scp,
                               &Bs[cur ^ 1][row * LS + scp]);
            }
            wait_async_le16();                  // current tile landed; next 16 in flight
        } else {
            wait_async_le0();
        }
        __syncthreads();

#pragma unroll
        for (int kk = 0; kk < 32; kk += 4) {
            v2f a[2], b[4];
#pragma unroll
            for (int i = 0; i < 2; ++i)
                a[i] = *(const v2f*)&As[cur][(waveM * 32 + 16 * i + lm) * LS + kk + 2 * lh];
#pragma unroll
            for (int j = 0; j < 4; ++j)
                b[j] = *(const v2f*)&Bs[cur][(waveN * 64 + 16 * j + lm) * LS + kk + 2 * lh];
#pragma unroll
            for (int i = 0; i < 2; ++i)
#pragma unroll
                for (int j = 0; j < 4; ++j)
                    acc[i][j] = __builtin_amdgcn_wmma_f32_16x16x4_f32(
                        false, a[i], false, b[j], (short)0, acc[i][j], false, false);
        }
        __syncthreads();                        // before next iteration overwrites cur^1
    }

    // C layout: VGPR r -> (M = r + 8*(lane>>4), N = lane&15)
#pragma unroll
    for (int i = 0; i < 2; ++i)
#pragma unroll
        for (int j = 0; j < 4; ++j)
#pragma unroll
            for (int r = 0; r < 8; ++r) {
                int row = m0 + waveM * 32 + 16 * i + r + 8 * lh;
                int col = n0 + waveN * 64 + 16 * j + lm;
                float v = acc[i][j][r];
                if (EPI) v = fmaf(Dv[col], U[(size_t)row * ldc + col], v);
                C[(size_t)row * ldc + col] = v;
            }
}

extern "C" void kernel_launch(void* const* d_in, const int* in_sizes, int n_in,
                              void* d_out, int out_size, void* d_ws, size_t ws_size,
                              hipStream_t stream) {
    (void)in_sizes; (void)n_in; (void)out_size; (void)ws_size;
    const float* u  = (const float*)d_in[0];   // (8,2048,1024)
    const float* ap = (const float*)d_in[1];   // (512,)
    const float* Bw = (const float*)d_in[2];   // (1024,1024)
    const float* Cw = (const float*)d_in[3];   // (1024,1024)
    const float* Dv = (const float*)d_in[4];   // (1024,)
    float* y = (float*)d_out;

    float* ws   = (float*)d_ws;
    float* X    = ws;                                   // MTOT*KX  (~72 MB)
    float* FB   = X  + (size_t)MTOT * KX;               // KX*DM
    float* Wc   = FB + (size_t)KX * DM;                 // DM*KX
    float* lre  = Wc + (size_t)DM * KX;                 // NKP
    float* lim  = lre + NKP;                            // NKP
    float* trig = lim + NKP;                            // 2048

    trig_kernel<<<4, 256, 0, stream>>>(trig);
    lambda_kernel<<<9, 64, 0, stream>>>(ap, trig, lre, lim);
    fb_kernel<<<KX / 8, 256, 0, stream>>>(Bw, trig, FB);
    cf_kernel<<<DM, 288, 0, stream>>>(Cw, trig, Wc);

    // GEMM1: X[m][k'] = sum_d u[m][d] * FB[k'][d]
    wmma_gemm<false><<<dim3(KX / 128, MTOT / 128), 256, 0, stream>>>(
        u, FB, X, DM, DM, DM, KX, nullptr, nullptr);

    // per-mode linear recurrence over time (in place)
    scan_kernel<<<(BATCH * NKP + 255) / 256, 256, 0, stream>>>(X, lre, lim);

    // GEMM2: y[m][d] = sum_k' X[m][k'] * Wc[d][k'] + D[d]*u[m][d]
    wmma_gemm<true><<<dim3(DM / 128, MTOT / 128), 256, 0, stream>>>(
        X, Wc, y, KX, KX, KX, DM, u, Dv);
}